// calc_padding_8641474200202
// MI455X (gfx1250) — compile-verified
//
#include <hip/hip_runtime.h>

// ---------------------------------------------------------------------------
// Problem: zero-pad x (8,128,256,256) to xp (8,128,260,260), then 8 sequential
// border passes, each a tiny 6->8->8->1 ReLU MLP slid along one border line.
//
// Roofline: ~545 MB of HBM traffic (write xp 277MB + read x 268MB) vs ~0.5
// GFLOP of MLP math -> purely memory bound (~23us at 23.3 TB/s). Structure:
//   1) one streaming pad-copy kernel (float4 stores)
//   2) eight small border kernels, ordered on the stream (true dependencies),
//      with the 8x6 / 8x8 MLP layers mapped onto V_WMMA_F32_16X16X4_F32
//      (fp32 WMMA, K split into K=4 tiles), 16 border positions per wave.
// ---------------------------------------------------------------------------

typedef float v2f __attribute__((ext_vector_type(2)));
typedef float v8f __attribute__((ext_vector_type(8)));

#define BATCH 8
#define CHN   128
#define HW    256
#define W2D   260          // padded side
#define IMG   (W2D * W2D)  // 67600 floats per (b,c) image
#define LOUT  258          // outputs per border line
#define TPL   17           // ceil(258/16) tiles of 16 positions per line

// ---------------------------------------------------------------------------
// Kernel 1: build xp = zero-padded x. One float4 store per thread.
// 65 quads per 260-wide row; interior loads are scalar (misaligned by 2 cols).
// ---------------------------------------------------------------------------
__global__ __launch_bounds__(256) void pad_copy_kernel(
    const float* __restrict__ x, float* __restrict__ xp)
{
    long idx = (long)blockIdx.x * blockDim.x + threadIdx.x;  // over B*C*260*65
    int  q   = (int)(idx % 65);
    long r   = idx / 65;
    int  row = (int)(r % W2D);
    long bc  = r / W2D;

    float4 v = make_float4(0.f, 0.f, 0.f, 0.f);
    if (row >= 2 && row < 258) {
        const float* xr = x + bc * (long)(HW * HW) + (long)(row - 2) * HW;
        int c0 = q * 4;
        float t[4];
#pragma unroll
        for (int j = 0; j < 4; ++j) {
            int c = c0 + j;
            t[j] = (c >= 2 && c < 258) ? xr[c - 2] : 0.f;
        }
        v = make_float4(t[0], t[1], t[2], t[3]);
    }
    float4* dst = (float4*)(xp + bc * (long)IMG + (long)row * W2D) + q;
    *dst = v;   // 16B-aligned: 67600, 260, 4q all multiples of 4 floats
}

// ---------------------------------------------------------------------------
// Kernel 2: one border pass. Each wave handles 16 consecutive positions.
//   line0/line1: the two gathered source lines (rows if isCol==0, cols else)
//   dstLine:     the border line written (coordinate 1+l along the border)
//   wsel:        which of the 4 stacked parameter sets (top/bot/left/right)
//
// WMMA f32 16x16x4 layouts used (wave32):
//   A (16x4): lane m   -> (A[m][0], A[m][1]) ; lane m+16 -> (A[m][2], A[m][3])
//   B (4x16): lane n   -> (B[0][n], B[1][n]) ; lane n+16 -> (B[2][n], B[3][n])
//   C/D:      VGPR j   -> row j (lanes 0-15, N=lane) / row j+8 (lanes 16-31)
// Weight rows >= 8 are zero, so D rows 8..15 are exactly 0.
// ---------------------------------------------------------------------------
__global__ __launch_bounds__(256) void border_mlp_kernel(
    float* __restrict__ xp,
    const float* __restrict__ w1, const float* __restrict__ b1,
    const float* __restrict__ w2, const float* __restrict__ b2,
    const float* __restrict__ w3, const float* __restrict__ b3,
    int wsel, int line0, int line1, int dstLine, int isCol)
{
    const int  lane = threadIdx.x & 31;
    const int  ln   = lane & 15;
    const bool hi   = lane >= 16;
    const int  wave = (int)(((long)blockIdx.x * blockDim.x + threadIdx.x) >> 5);

    const int  tline = wave % TPL;          // tile within the border line
    const int  bc    = wave / TPL;          // (b,c) image index
    float* img = xp + (long)bc * IMG;

    const int  l  = tline * 16 + ln;        // output position 0..257 (+tail)
    const int  lc = (l < LOUT) ? l : (LOUT - 1);   // clamp tail lanes' loads
    const long sS = isCol ? (long)W2D : 1;         // stride along the border
    const long o0 = isCol ? (long)line0 : (long)line0 * W2D;
    const long o1 = isCol ? (long)line1 : (long)line1 * W2D;

    // ---- Layer 1 inputs: 6x16 matrix, K = r*3+k, split K=(0..3) + (4..5,0,0)
    v2f B1t0, B1t1;
    if (!hi) {
        B1t0.x = img[o0 + (long)(lc + 0) * sS];    // K0 = (r0,k0)
        B1t0.y = img[o0 + (long)(lc + 1) * sS];    // K1 = (r0,k1)
        B1t1.x = img[o1 + (long)(lc + 1) * sS];    // K4 = (r1,k1)
        B1t1.y = img[o1 + (long)(lc + 2) * sS];    // K5 = (r1,k2)
    } else {
        B1t0.x = img[o0 + (long)(lc + 2) * sS];    // K2 = (r0,k2)
        B1t0.y = img[o1 + (long)(lc + 0) * sS];    // K3 = (r1,k0)
        B1t1.x = 0.f;                              // K6 pad
        B1t1.y = 0.f;                              // K7 pad
    }

    // ---- Layer 1 weights as A tiles (w1[wsel][h][r][k], K flattened = r*3+k)
    v2f A1t0 = {0.f, 0.f}, A1t1 = {0.f, 0.f};
    if (ln < 8) {
        const float* W1r = w1 + wsel * 48 + ln * 6;
        if (!hi) { A1t0.x = W1r[0]; A1t0.y = W1r[1]; A1t1.x = W1r[4]; A1t1.y = W1r[5]; }
        else     { A1t0.x = W1r[2]; A1t0.y = W1r[3]; }
    }

    // ---- C = bias broadcast (rows 8..15 zero)
    v8f acc;
    const float* B1b = b1 + wsel * 8;
#pragma unroll
    for (int j = 0; j < 8; ++j) acc[j] = hi ? 0.f : B1b[j];

    acc = __builtin_amdgcn_wmma_f32_16x16x4_f32(false, A1t0, false, B1t0,
                                                (short)0, acc, false, false);
    acc = __builtin_amdgcn_wmma_f32_16x16x4_f32(false, A1t1, false, B1t1,
                                                (short)0, acc, false, false);
#pragma unroll
    for (int j = 0; j < 8; ++j) acc[j] = fmaxf(acc[j], 0.f);   // ReLU

    // ---- D-layout -> B-layout for layer 2: h1[j][n] lives in VGPR j, lanes
    // 0..15. Broadcast across the wave, hi half selects rows 2,3 / 6,7.
    float s0 = __shfl(acc[0], ln, 32), s1 = __shfl(acc[1], ln, 32);
    float s2 = __shfl(acc[2], ln, 32), s3 = __shfl(acc[3], ln, 32);
    float s4 = __shfl(acc[4], ln, 32), s5 = __shfl(acc[5], ln, 32);
    float s6 = __shfl(acc[6], ln, 32), s7 = __shfl(acc[7], ln, 32);
    v2f B2t0, B2t1;
    B2t0.x = hi ? s2 : s0;  B2t0.y = hi ? s3 : s1;
    B2t1.x = hi ? s6 : s4;  B2t1.y = hi ? s7 : s5;

    // ---- Layer 2 weights: w2[wsel][g][h], A[g][h], K=h split 0..3 / 4..7
    v2f A2t0 = {0.f, 0.f}, A2t1 = {0.f, 0.f};
    if (ln < 8) {
        const float* W2r = w2 + wsel * 64 + ln * 8;
        if (!hi) { A2t0.x = W2r[0]; A2t0.y = W2r[1]; A2t1.x = W2r[4]; A2t1.y = W2r[5]; }
        else     { A2t0.x = W2r[2]; A2t0.y = W2r[3]; A2t1.x = W2r[6]; A2t1.y = W2r[7]; }
    }
    v8f acc2;
    const float* B2b = b2 + wsel * 8;
#pragma unroll
    for (int j = 0; j < 8; ++j) acc2[j] = hi ? 0.f : B2b[j];

    acc2 = __builtin_amdgcn_wmma_f32_16x16x4_f32(false, A2t0, false, B2t0,
                                                 (short)0, acc2, false, false);
    acc2 = __builtin_amdgcn_wmma_f32_16x16x4_f32(false, A2t1, false, B2t1,
                                                 (short)0, acc2, false, false);

    // ---- Layer 3: o[n] = relu(b3 + sum_g w3[g]*relu(h2[g][n])). Column n of
    // h2 sits across acc2[0..7] in lane n (n<16) -> 8 per-lane FMAs.
    const float* W3r = w3 + wsel * 8;
    float o = b3[wsel];
#pragma unroll
    for (int g = 0; g < 8; ++g) o += W3r[g] * fmaxf(acc2[g], 0.f);
    o = fmaxf(o, 0.f);

    if (!hi && l < LOUT) {
        long d = isCol ? ((long)(1 + l) * W2D + dstLine)
                       : ((long)dstLine * W2D + (1 + l));
        img[d] = o;
    }
}

// ---------------------------------------------------------------------------
// Launch: pad-copy, then the 8 border passes in the reference's exact order
// (top, bottom, left, right for i=0 then i=1) — stream order carries the
// true data dependencies between passes.
// ---------------------------------------------------------------------------
extern "C" void kernel_launch(void* const* d_in, const int* in_sizes, int n_in,
                              void* d_out, int out_size, void* d_ws, size_t ws_size,
                              hipStream_t stream)
{
    const float* x  = (const float*)d_in[0];
    const float* w1 = (const float*)d_in[1];
    const float* b1 = (const float*)d_in[2];
    const float* w2 = (const float*)d_in[3];
    const float* b2 = (const float*)d_in[4];
    const float* w3 = (const float*)d_in[5];
    const float* b3 = (const float*)d_in[6];
    float* xp = (float*)d_out;   // output IS xp; compute in place

    // pad copy: B*C*260*65 float4-quads, exact grid
    {
        long threads = (long)BATCH * CHN * W2D * 65;       // 17,305,600
        int  blocks  = (int)(threads / 256);               // 67,600 exact
        pad_copy_kernel<<<blocks, 256, 0, stream>>>(x, xp);
    }

    // border passes: B*C*TPL waves, 32 threads each, exact grid
    const int bblocks = (BATCH * CHN * TPL * 32) / 256;    // 2,176 exact
    for (int i = 0; i < 2; ++i) {
        // top:    rows (2-i, 3-i) -> row 1-i
        border_mlp_kernel<<<bblocks, 256, 0, stream>>>(xp, w1, b1, w2, b2, w3, b3,
                                                       0, 2 - i, 3 - i, 1 - i, 0);
        // bottom: rows (255+i, 256+i) -> row 258+i
        border_mlp_kernel<<<bblocks, 256, 0, stream>>>(xp, w1, b1, w2, b2, w3, b3,
                                                       1, 255 + i, 256 + i, 258 + i, 0);
        // left:   cols (2-i, 3-i) -> col 1-i
        border_mlp_kernel<<<bblocks, 256, 0, stream>>>(xp, w1, b1, w2, b2, w3, b3,
                                                       2, 2 - i, 3 - i, 1 - i, 1);
        // right:  cols (255+i, 256+i) -> col 258+i
        border_mlp_kernel<<<bblocks, 256, 0, stream>>>(xp, w1, b1, w2, b2, w3, b3,
                                                       3, 255 + i, 256 + i, 258 + i, 1);
    }
}